// Net_MMIF_40278203302343
// MI455X (gfx1250) — compile-verified
//
#include <hip/hip_runtime.h>

// Problem constants (from the reference)
#define D3c 33
#define D2c 33
#define D1c 256
#define LOGW 11
#define WMASK 2047
#define HWc (2048 * 2048)        // H*W = 4,194,304 = 2^22
#define LOGHW 22
#define NPIX (2 * HWc)           // B*H*W = 8,388,608
#define LUT3_N (D3c * D3c * D3c) // 35937 floats = 143,748 B  (fits 320KB WGP LDS)
#define LUT2_N (D2c * D2c)       // 1089 floats

typedef __attribute__((ext_vector_type(4))) unsigned uint4v;
typedef __attribute__((ext_vector_type(8))) unsigned uint8v;

// ---------------------------------------------------------------------------
// LDS byte offset of a shared-memory pointer (flat shared address low 32 bits
// == workgroup-relative LDS byte offset).
// ---------------------------------------------------------------------------
__device__ __forceinline__ unsigned lds_addr_of(const void* p) {
    return (unsigned)(unsigned long long)p;
}

// ---------------------------------------------------------------------------
// Tensor Data Mover: one instruction DMAs a 1-D run of `nelem` floats from
// global memory into LDS.  D# per CDNA5 ISA: group0 = {count=1, lds_addr,
// global_addr lo, global_addr hi | type=2}, group1 = {data_size=4B,
// tensor_dim0=N, tensor_dim1=1, tile_dim0=N, tile_dim1/2 unused, stride0=N}.
// Groups 2/3 omitted (both NULL -> tensor up to 2D).  TENSORcnt-tracked.
// Caller must be a single wave (TDM ignores EXEC) and follow with
// s_wait_tensorcnt + workgroup barrier.
// ---------------------------------------------------------------------------
__device__ __forceinline__ void tdm_load_1d_to_lds(const float* g, unsigned lds_off,
                                                   unsigned nelem) {
    unsigned long long ga = (unsigned long long)g;
    uint4v g0 = { 1u,                                    // count=1, user desc
                  lds_off,                               // LDS byte address
                  (unsigned)ga,                          // global_addr[31:0]
                  (unsigned)((ga >> 32) & 0x01FFFFFFull) // global_addr[56:32]
                      | 0x80000000u };                   // type=2 ("image")
    uint8v g1 = { 0x2u << 16,                 // data_size=2 (4 bytes/elem)
                  (nelem & 0xFFFFu) << 16,    // tensor_dim0[15:0]
                  (nelem >> 16) | (1u << 16), // tensor_dim0[31:16] | tensor_dim1=1
                  (nelem & 0xFFFFu) << 16,    // tensor_dim1 hi=0 | tile_dim0
                  0u,                         // tile_dim1=0, tile_dim2=0 (unused)
                  nelem,                      // tensor_dim0_stride[31:0]
                  0u, 0u };
    asm volatile("tensor_load_to_lds %0, %1"
                 :: "s"(g0), "s"(g1) : "memory");
}

// ---------------------------------------------------------------------------
// LUT math helpers
// ---------------------------------------------------------------------------
__device__ __forceinline__ float clip01(float x) { return fminf(fmaxf(x, 0.0f), 1.0f); }

__device__ __forceinline__ float bilin33(const float* t, float u, float v) {
    float us = clip01(u) * 32.0f, vs = clip01(v) * 32.0f;
    int i = min((int)us, 31), j = min((int)vs, 31);
    float fi = us - (float)i, fj = vs - (float)j;
    const float* p = t + i * 33 + j;
    float c00 = p[0], c01 = p[1], c10 = p[33], c11 = p[34];
    float a = c00 + fj * (c01 - c00);
    float b = c10 + fj * (c11 - c10);
    return clip01(a + fi * (b - a));
}

__device__ __forceinline__ float trilin33(const float* t, float u, float v, float w) {
    float us = clip01(u) * 32.0f, vs = clip01(v) * 32.0f, ws = clip01(w) * 32.0f;
    int i = min((int)us, 31), j = min((int)vs, 31), k = min((int)ws, 31);
    float fi = us - (float)i, fj = vs - (float)j, fk = ws - (float)k;
    const float* p = t + (i * 33 + j) * 33 + k;   // 8 LDS gathers
    float c000 = p[0],    c001 = p[1];
    float c010 = p[33],   c011 = p[34];
    float c100 = p[1089], c101 = p[1090];
    float c110 = p[1122], c111 = p[1123];
    float c00 = c000 + fk * (c001 - c000);
    float c01 = c010 + fk * (c011 - c010);
    float c10 = c100 + fk * (c101 - c100);
    float c11 = c110 + fk * (c111 - c110);
    float c0 = c00 + fj * (c01 - c00);
    float c1 = c10 + fj * (c11 - c10);
    return clip01(c0 + fi * (c1 - c0));
}

__device__ __forceinline__ float lut1d256(const float* t, float x) {
    float xs = clip01(x) * 255.0f;
    int i = min((int)xs, 254);
    float f = xs - (float)i;
    return t[i] + f * (t[i + 1] - t[i]);
}

// ---------------------------------------------------------------------------
// Kernel 1: pg0 = clip(lut2d(A, B))   (float4-vectorized, TDM-staged LUT)
// ---------------------------------------------------------------------------
__global__ __launch_bounds__(256) void k_lut2d(const float4* __restrict__ A,
                                               const float4* __restrict__ B,
                                               const float* __restrict__ lut,
                                               float4* __restrict__ out) {
    __shared__ float t[LUT2_N];
    if ((threadIdx.x >> 5) == 0) {                 // wave 0 only
        tdm_load_1d_to_lds(lut, lds_addr_of(t), LUT2_N);
        __builtin_amdgcn_s_wait_tensorcnt(0);
    }
    __syncthreads();

    int p = blockIdx.x * 256 + threadIdx.x;
    float4 a = A[p], b = B[p];
    float4 r;
    r.x = bilin33(t, a.x, b.x);
    r.y = bilin33(t, a.y, b.y);
    r.z = bilin33(t, a.z, b.z);
    r.w = bilin33(t, a.w, b.w);
    out[p] = r;
}

// ---------------------------------------------------------------------------
// Kernels 2-5: spatial-detail 3D LUT pass.
// n1[y][x] = in[(y-dy)&2047][(x-dx)&2047], n2 uses 2*dy, 2*dx (jnp.roll).
// Wave 0 TDM-stages the 143KB 33^3 LUT into LDS in a single instruction; the
// block then processes 4096 contiguous pixels (16 coalesced sweeps of 256
// lanes).  143,748 B LDS -> 2 blocks resident per 320KB WGP.
// ---------------------------------------------------------------------------
__global__ __launch_bounds__(256) void k_sdlut(const float* __restrict__ in,
                                               const float* __restrict__ lut,
                                               float* __restrict__ out,
                                               int dy, int dx) {
    __shared__ float t[LUT3_N];
    if ((threadIdx.x >> 5) == 0) {                 // wave 0 only (TDM ignores EXEC)
        tdm_load_1d_to_lds(lut, lds_addr_of(t), LUT3_N);
        __builtin_amdgcn_s_wait_tensorcnt(0);
    }
    __syncthreads();

    const int base = blockIdx.x * 4096;
#pragma unroll
    for (int k = 0; k < 16; ++k) {
        int p = base + (k << 8) + threadIdx.x;
        int b   = p >> LOGHW;
        int rem = p & (HWc - 1);
        int y = rem >> LOGW, x = rem & WMASK;
        float c = in[p];
        int y1 = (y - dy) & WMASK,     x1 = (x - dx) & WMASK;
        int y2 = (y - 2 * dy) & WMASK, x2 = (x - 2 * dx) & WMASK;
        int bb = b << LOGHW;
        float n1 = in[bb + (y1 << LOGW) + x1];
        float n2 = in[bb + (y2 << LOGW) + x2];
        out[p] = trilin33(t, c, n1, n2);
    }
}

// ---------------------------------------------------------------------------
// Kernel 6: 1D LUTs (pgf/cb/cr, three TDM loads) + YCC->RGB, float4 stores
// ---------------------------------------------------------------------------
__global__ __launch_bounds__(256) void k_final(const float4* __restrict__ sd,
                                               const float4* __restrict__ cbi,
                                               const float4* __restrict__ cri,
                                               const float* __restrict__ lpg,
                                               const float* __restrict__ lcb,
                                               const float* __restrict__ lcr,
                                               float* __restrict__ out) {
    __shared__ float t[3 * D1c];
    if ((threadIdx.x >> 5) == 0) {                 // wave 0 only
        tdm_load_1d_to_lds(lpg, lds_addr_of(t),       D1c);
        tdm_load_1d_to_lds(lcb, lds_addr_of(t + 256), D1c);
        tdm_load_1d_to_lds(lcr, lds_addr_of(t + 512), D1c);
        __builtin_amdgcn_s_wait_tensorcnt(0);
    }
    __syncthreads();

    int p = blockIdx.x * 256 + threadIdx.x;   // float4 index
    int pix = p << 2;
    int b   = pix >> LOGHW;
    int rem = pix & (HWc - 1);
    float4 s = sd[p], u = cbi[p], v = cri[p];

    float y0 = lut1d256(t, s.x),        y1 = lut1d256(t, s.y);
    float y2 = lut1d256(t, s.z),        y3 = lut1d256(t, s.w);
    float b0 = lut1d256(t + 256, u.x) - 0.5f, b1 = lut1d256(t + 256, u.y) - 0.5f;
    float b2 = lut1d256(t + 256, u.z) - 0.5f, b3 = lut1d256(t + 256, u.w) - 0.5f;
    float r0 = lut1d256(t + 512, v.x) - 0.5f, r1 = lut1d256(t + 512, v.y) - 0.5f;
    float r2 = lut1d256(t + 512, v.z) - 0.5f, r3 = lut1d256(t + 512, v.w) - 0.5f;

    float* o = out + b * 3 * HWc + rem;
    float4 c0, c1, c2;
    c0.x = y0 + r0 * 1.402f; c0.y = y1 + r1 * 1.402f;
    c0.z = y2 + r2 * 1.402f; c0.w = y3 + r3 * 1.402f;
    c1.x = y0 - b0 * 0.344136f - r0 * 0.714136f;
    c1.y = y1 - b1 * 0.344136f - r1 * 0.714136f;
    c1.z = y2 - b2 * 0.344136f - r2 * 0.714136f;
    c1.w = y3 - b3 * 0.344136f - r3 * 0.714136f;
    c2.x = y0 + b0 * 1.772f; c2.y = y1 + b1 * 1.772f;
    c2.z = y2 + b2 * 1.772f; c2.w = y3 + b3 * 1.772f;
    *(float4*)(o)           = c0;
    *(float4*)(o + HWc)     = c1;
    *(float4*)(o + 2 * HWc) = c2;
}

// ---------------------------------------------------------------------------
// Launch: pg0 -> 4 ping-pong sdlut passes -> final combine.
// bufA = d_ws (32MB).  bufB = last 32MB of d_out (96MB total) — safe because
// the final kernel only reads bufA/cb/cr and overwrites all of d_out.
// ---------------------------------------------------------------------------
extern "C" void kernel_launch(void* const* d_in, const int* in_sizes, int n_in,
                              void* d_out, int out_size, void* d_ws, size_t ws_size,
                              hipStream_t stream) {
    (void)in_sizes; (void)n_in; (void)out_size; (void)ws_size;
    const float* A    = (const float*)d_in[0];
    const float* Bi   = (const float*)d_in[1];
    const float* cb   = (const float*)d_in[2];
    const float* cr   = (const float*)d_in[3];
    const float* l2   = (const float*)d_in[4];
    const float* l3_0   = (const float*)d_in[5];
    const float* l3_90  = (const float*)d_in[6];
    const float* l3_180 = (const float*)d_in[7];
    const float* l3_270 = (const float*)d_in[8];
    const float* l1pg = (const float*)d_in[9];
    const float* l1cb = (const float*)d_in[10];
    const float* l1cr = (const float*)d_in[11];

    float* bufA = (float*)d_ws;                 // NPIX floats (32 MB)
    float* bufB = (float*)d_out + 4 * HWc;      // last NPIX floats of d_out

    dim3 blk(256);
    k_lut2d<<<NPIX / 1024, blk, 0, stream>>>((const float4*)A, (const float4*)Bi,
                                             l2, (float4*)bufA);
    k_sdlut<<<NPIX / 4096, blk, 0, stream>>>(bufA, l3_0,   bufB,  0, -1);
    k_sdlut<<<NPIX / 4096, blk, 0, stream>>>(bufB, l3_90,  bufA,  1,  0);
    k_sdlut<<<NPIX / 4096, blk, 0, stream>>>(bufA, l3_180, bufB,  0,  1);
    k_sdlut<<<NPIX / 4096, blk, 0, stream>>>(bufB, l3_270, bufA, -1,  0);
    k_final<<<NPIX / 1024, blk, 0, stream>>>((const float4*)bufA, (const float4*)cb,
                                             (const float4*)cr, l1pg, l1cb, l1cr,
                                             (float*)d_out);
}